// MAMConv1d_60155311948479
// MI455X (gfx1250) — compile-verified
//
#include <hip/hip_runtime.h>
#include <math.h>

// ---------------------------------------------------------------------------
// MAMConv1d: y[b,o,l] = max_{c,k}(x[b,c,l+k]*w[o,c,k]) + min_{c,k}(...) + bias[o]
//   x: (8,64,1024) f32, w: (64,64,3) f32, bias: (64) f32, y: (8,64,1022) f32
// Max+min reduction => WMMA not applicable. CDNA5 path used: TDM
// tensor_load_to_lds staging + s_wait_tensorcnt, wave32 layout.
// ---------------------------------------------------------------------------

#define AS3 __attribute__((address_space(3)))

typedef unsigned int u32x4 __attribute__((ext_vector_type(4)));
typedef int          i32x4 __attribute__((ext_vector_type(4)));
typedef int          i32x8 __attribute__((ext_vector_type(8)));

#if __has_builtin(__builtin_amdgcn_tensor_load_to_lds) && \
    __has_builtin(__builtin_amdgcn_s_wait_tensorcnt)
#define USE_TDM 1
#else
#define USE_TDM 0
#endif

constexpr int Bn   = 8;
constexpr int Cn   = 64;
constexpr int Ln   = 1024;
constexpr int On   = 64;
constexpr int Kn   = 3;
constexpr int LOUT = 1022;          // (1024 - 3) + 1
constexpr int TL   = 146;           // 7 * 146 == 1022, tiles never read x OOB
constexpr int XS   = TL + Kn - 1;   // 148 floats per channel row in LDS
constexpr int NW   = Cn * Kn;       // 192 weights per output channel

__device__ __forceinline__ unsigned ldsOffsetOf(const void* p) {
    // generic -> LDS addrspace cast; 32-bit local pointer value is the byte
    // offset within the workgroup's LDS window (what D#.lds_addr expects).
    return (unsigned)(unsigned long long)(AS3 const char*)p;
}

#if USE_TDM
// Issue one TDM 2-D tile load: tile (tile_d0 x tile_d1) elements of 4 bytes
// from a row-major tensor with row stride stride0 (elements) into contiguous
// LDS starting at lds_byte_addr. D# packing per CDNA5 ISA ch.8.
__device__ __forceinline__ void tdm_load_2d(unsigned lds_byte_addr,
                                            unsigned long long gaddr,
                                            unsigned tensor_d0, unsigned tensor_d1,
                                            unsigned tile_d0,   unsigned tile_d1,
                                            unsigned long long stride0) {
    u32x4 g0;
    g0.x = 1u;                                            // count=1, user descriptor
    g0.y = lds_byte_addr;                                 // lds_addr (bytes)
    g0.z = (unsigned)(gaddr & 0xFFFFFFFFull);             // global_addr[31:0]
    g0.w = ((unsigned)(gaddr >> 32) & 0x01FFFFFFu)        // global_addr[56:32]
           | 0x80000000u;                                 // type=2 ("image")

    i32x8 g1;
    g1[0] = (int)(2u << 16);                              // data_size=2 (4B), mask=0
    g1[1] = (int)((tensor_d0 & 0xFFFFu) << 16);           // tensor_dim0[15:0]
    g1[2] = (int)((tensor_d0 >> 16) |
                  ((tensor_d1 & 0xFFFFu) << 16));         // dim0 hi | dim1 lo
    g1[3] = (int)((tensor_d1 >> 16) |
                  ((tile_d0 & 0xFFFFu) << 16));           // dim1 hi | tile_dim0
    g1[4] = (int)(tile_d1 & 0xFFFFu);                     // tile_dim1 (tile_dim2=0)
    g1[5] = (int)(unsigned)(stride0 & 0xFFFFFFFFull);     // tensor_dim0_stride lo32
    g1[6] = (int)(unsigned)((stride0 >> 32) & 0xFFFFull); // stride hi16 (dim1_stride=0)
    g1[7] = 0;

    i32x4 g2 = {0, 0, 0, 0};
    i32x4 g3 = {0, 0, 0, 0};
#if __clang_major__ >= 23
    i32x8 g4 = {0, 0, 0, 0, 0, 0, 0, 0};
    __builtin_amdgcn_tensor_load_to_lds(g0, g1, g2, g3, g4, 0);
#else
    __builtin_amdgcn_tensor_load_to_lds(g0, g1, g2, g3, 0);
#endif
}
#endif // USE_TDM

__global__ __launch_bounds__(256)
void MAMConv1d_kernel(const float* __restrict__ x,
                      const float* __restrict__ w,
                      const float* __restrict__ bias,
                      float* __restrict__ y) {
    __shared__ float xs[Cn * XS];   // x tile: 64 rows x 148 (37.9 KB)
    __shared__ float ws[On * NW];   // full weights, (o, c*3+k) layout (48 KB)

    const int tile = blockIdx.x;    // 0..6
    const int b    = blockIdx.y;    // 0..7
    const int l0   = tile * TL;

#if USE_TDM
    if (threadIdx.x == 0) {
        // x tile: rows c=0..63, columns l0..l0+147, row stride L
        const unsigned long long xg =
            (unsigned long long)(const void*)x +
            4ull * (unsigned long long)((b * Cn) * Ln + l0);
        tdm_load_2d(ldsOffsetOf(xs), xg, Ln, Cn, XS, Cn, Ln);
        // weights: flat contiguous 12288-element copy
        const unsigned long long wg = (unsigned long long)(const void*)w;
        tdm_load_2d(ldsOffsetOf(ws), wg, (unsigned)(On * NW), 1u,
                    (unsigned)(On * NW), 1u, (unsigned long long)(On * NW));
    }
    // All waves drain TENSORcnt (0 outstanding for non-issuing waves; robust
    // even if the compiler does not branch around the EXEC=0 path above).
    __builtin_amdgcn_s_wait_tensorcnt(0);
#else
    for (int i = threadIdx.x; i < Cn * XS; i += 256) {
        const int c = i / XS, t = i - c * XS;
        xs[i] = x[(b * Cn + c) * Ln + l0 + t];
    }
    for (int i = threadIdx.x; i < On * NW; i += 256) ws[i] = w[i];
#endif
    __syncthreads();

    const int lane = threadIdx.x & 31;   // consecutive l -> conflict-free LDS
    const int wave = threadIdx.x >> 5;   // 8 waves, each owns 8 output channels

    for (int oi = 0; oi < On / 8; ++oi) {
        const int o = wave * 8 + oi;           // uniform per wave -> ws broadcast
        const float* wr = &ws[o * NW];

        float mx[5], mn[5];
#pragma unroll
        for (int j = 0; j < 5; ++j) { mx[j] = -__builtin_inff(); mn[j] = __builtin_inff(); }

        for (int c = 0; c < Cn; ++c) {
            const float* xr = &xs[c * XS + lane];
#pragma unroll
            for (int k = 0; k < Kn; ++k) {
                const float wv = wr[c * Kn + k];   // LDS broadcast (same addr/wave)
#pragma unroll
                for (int j = 0; j < 5; ++j) {
                    const float p = xr[j * 32 + k] * wv;
                    mx[j] = fmaxf(mx[j], p);
                    mn[j] = fminf(mn[j], p);
                }
            }
        }

        const float bo = bias[o];
        float* yr = &y[(b * On + o) * LOUT + l0];
#pragma unroll
        for (int j = 0; j < 5; ++j) {
            const int ll = lane + j * 32;
            if (ll < TL) yr[ll] = mx[j] + mn[j] + bo;  // 146 = 4*32 + 18 tail
        }
    }
}

extern "C" void kernel_launch(void* const* d_in, const int* in_sizes, int n_in,
                              void* d_out, int out_size, void* d_ws, size_t ws_size,
                              hipStream_t stream) {
    (void)in_sizes; (void)n_in; (void)d_ws; (void)ws_size; (void)out_size;
    const float* x    = (const float*)d_in[0];
    const float* w    = (const float*)d_in[1];
    const float* bias = (const float*)d_in[2];
    // d_in[3] is stride == 1 (baked into the kernel).
    float* y = (float*)d_out;

    dim3 grid(LOUT / TL, Bn);  // (7, 8) blocks, 256 threads (8 wave32s) each
    MAMConv1d_kernel<<<grid, dim3(256), 0, stream>>>(x, w, bias, y);
}